// DLI_loss_1_6614249636351
// MI455X (gfx1250) — compile-verified
//
#include <hip/hip_runtime.h>

typedef float v2f __attribute__((ext_vector_type(2)));
typedef float v8f __attribute__((ext_vector_type(8)));

#define BB 32
#define SS 4096
#define DD 512
#define TT 64

// ---------------------------------------------------------------------------
// Kernel 1: segmented mean over encoder_output rows.
// grid = B*T blocks, 256 threads; each thread owns a float2 column strip.
// Streams 256 MiB once -> non-temporal loads (TH=NT) to avoid cache pollution.
// ---------------------------------------------------------------------------
__global__ void __launch_bounds__(256)
seg_mean_kernel(const float* __restrict__ enc,
                const int* __restrict__ ids,
                float* __restrict__ states) {
    int blk = blockIdx.x;            // b*T + t
    int b   = blk >> 6;
    int t   = blk & 63;
    int end   = ids[b * TT + t];
    int start = (t == 0) ? 0 : (ids[b * TT + t - 1] + 1);
    int tid = threadIdx.x;           // 0..255 -> columns [2*tid, 2*tid+1]

    const v2f* base = (const v2f*)(enc + (size_t)b * SS * DD) + tid;
    v2f acc = {0.0f, 0.0f};
    for (int s = start; s <= end; ++s) {
        v2f v = __builtin_nontemporal_load(base + (size_t)s * (DD / 2));
        acc += v;
    }
    float inv = 1.0f / (float)(end - start + 1);
    acc *= inv;
    ((v2f*)(states + (size_t)blk * DD))[tid] = acc;
}

// ---------------------------------------------------------------------------
// Kernel 0: expand W (1024 x 2, row-major) into a zero-padded 512 x 16 B tile:
// col 0,1 = Wl columns; col 2,3 = Wr columns; cols 4..15 = 0.
// ---------------------------------------------------------------------------
__global__ void __launch_bounds__(256)
build_wcat_kernel(const float* __restrict__ W, float* __restrict__ wcat) {
    int k = blockIdx.x * blockDim.x + threadIdx.x;   // 0..511
    if (k < DD) {
        float* o = wcat + k * 16;
        o[0] = W[k * 2 + 0];
        o[1] = W[k * 2 + 1];
        o[2] = W[(DD + k) * 2 + 0];
        o[3] = W[(DD + k) * 2 + 1];
#pragma unroll
        for (int n = 4; n < 16; ++n) o[n] = 0.0f;
    }
}

// ---------------------------------------------------------------------------
// Kernel 2: scores(2048x4) = states(2048x512) @ wcat(512x16) via
// V_WMMA_F32_16X16X4_F32. One wave per 16-row tile; K accumulated over 128
// K=4 steps with two interleaved accumulators to break the RAW chain.
// A layout (16x4 f32): lane L holds row L&15; VGPR0 = K(2*(L>>4)),
//                      VGPR1 = K(2*(L>>4)+1).
// B layout (4x16 f32): VGPR0: lanes0-15 row K, lanes16-31 row K+2; VGPR1: +1.
// D layout: VGPR v -> row v + 8*(L>>4), col L&15.
// ---------------------------------------------------------------------------
__global__ void __launch_bounds__(32)
gemm_wmma_kernel(const float* __restrict__ states,
                 const float* __restrict__ wcat,
                 float* __restrict__ scores) {
    int tile  = blockIdx.x;          // 0..127 (16 rows each)
    int lane  = threadIdx.x;         // 0..31
    int rowlo = lane & 15;
    int khalf = lane >> 4;           // 0 or 1

    const float* arow  = states + ((size_t)tile * 16 + rowlo) * DD + 2 * khalf;
    const float* bbase = wcat + (2 * khalf) * 16 + rowlo;

    v8f acc0 = {};
    v8f acc1 = {};
    for (int k0 = 0; k0 < DD; k0 += 8) {
        v2f a0; a0.x = arow[k0];     a0.y = arow[k0 + 1];
        v2f b0; b0.x = bbase[(k0) * 16];     b0.y = bbase[(k0 + 1) * 16];
        acc0 = __builtin_amdgcn_wmma_f32_16x16x4_f32(
            /*neg_a=*/false, a0, /*neg_b=*/false, b0,
            /*c_mod=*/(short)0, acc0, /*reuse_a=*/false, /*reuse_b=*/false);

        v2f a1; a1.x = arow[k0 + 4]; a1.y = arow[k0 + 5];
        v2f b1; b1.x = bbase[(k0 + 4) * 16]; b1.y = bbase[(k0 + 5) * 16];
        acc1 = __builtin_amdgcn_wmma_f32_16x16x4_f32(
            false, a1, false, b1, (short)0, acc1, false, false);
    }
    v8f acc = acc0 + acc1;

    int n = lane & 15;
    if (n < 4) {
#pragma unroll
        for (int v = 0; v < 8; ++v) {
            int m = v + 8 * khalf;
            scores[((size_t)tile * 16 + m) * 4 + n] = acc[v];
        }
    }
}

// ---------------------------------------------------------------------------
// Kernel 3: per-batch pairwise NLL over the k<j triangle.
// logits[c] = lj[j][c] + rk[k][c] + bias[c]; target class = (k==j-1) ? 1 : 0
// nll = logsumexp(z0,z1) - z_target. LDS-reduce to partial[b].
// ---------------------------------------------------------------------------
__global__ void __launch_bounds__(256)
pair_loss_kernel(const float* __restrict__ scores,
                 const float* __restrict__ bias,
                 float* __restrict__ partial) {
    int b   = blockIdx.x;
    int tid = threadIdx.x;
    float bias0 = bias[0], bias1 = bias[1];
    const float* sb = scores + (size_t)b * TT * 4;

    float sum = 0.0f;
    for (int idx = tid; idx < TT * TT; idx += 256) {
        int j = idx >> 6;
        int k = idx & 63;
        if (k < j) {
            float z0 = sb[j * 4 + 0] + sb[k * 4 + 2] + bias0;
            float z1 = sb[j * 4 + 1] + sb[k * 4 + 3] + bias1;
            float m  = fmaxf(z0, z1);
            float lse = m + logf(expf(z0 - m) + expf(z1 - m));
            float zt  = (k == j - 1) ? z1 : z0;
            sum += lse - zt;
        }
    }

    __shared__ float red[256];
    red[tid] = sum;
    __syncthreads();
#pragma unroll
    for (int s = 128; s > 0; s >>= 1) {
        if (tid < s) red[tid] += red[tid + s];
        __syncthreads();
    }
    if (tid == 0) partial[b] = red[0];
}

// ---------------------------------------------------------------------------
// Kernel 4: reduce 32 per-batch partials, scale by 1/n_pairs, write d_out.
// ---------------------------------------------------------------------------
__global__ void __launch_bounds__(32)
finalize_kernel(const float* __restrict__ partial, float* __restrict__ out) {
    __shared__ float red[32];
    int tid = threadIdx.x;
    red[tid] = partial[tid];
    __syncthreads();
#pragma unroll
    for (int s = 16; s > 0; s >>= 1) {
        if (tid < s) red[tid] += red[tid + s];
        __syncthreads();
    }
    if (tid == 0) out[0] = red[0] / 64512.0f;   // B * T*(T-1)/2 = 32*2016
}

// ---------------------------------------------------------------------------
// Launcher
// ---------------------------------------------------------------------------
extern "C" void kernel_launch(void* const* d_in, const int* in_sizes, int n_in,
                              void* d_out, int out_size, void* d_ws, size_t ws_size,
                              hipStream_t stream) {
    const float* enc  = (const float*)d_in[0];   // (32, 4096, 512) f32
    const float* W    = (const float*)d_in[1];   // (1024, 2) f32
    const float* bias = (const float*)d_in[2];   // (2,) f32
    const int*   ids  = (const int*)d_in[3];     // (32, 64) i32
    float* out        = (float*)d_out;

    char* ws = (char*)d_ws;
    float* states  = (float*)(ws);                              // 2048*512*4 = 4 MiB
    float* wcat    = (float*)(ws + (size_t)BB * TT * DD * 4);   // 512*16*4  = 32 KiB
    float* scores  = (float*)((char*)wcat + (size_t)DD * 16 * 4); // 2048*4*4 = 32 KiB
    float* partial = (float*)((char*)scores + (size_t)BB * TT * 4 * 4); // 32 floats

    seg_mean_kernel<<<BB * TT, 256, 0, stream>>>(enc, ids, states);
    build_wcat_kernel<<<2, 256, 0, stream>>>(W, wcat);
    gemm_wmma_kernel<<<(BB * TT) / 16, 32, 0, stream>>>(states, wcat, scores);
    pair_loss_kernel<<<BB, 256, 0, stream>>>(scores, bias, partial);
    finalize_kernel<<<1, 32, 0, stream>>>(partial, out);
}